// Filter_41059887350001
// MI455X (gfx1250) — compile-verified
//
#include <hip/hip_runtime.h>

// Problem constants (from reference setup_inputs)
#define N      2048
#define N2     (2 * N)
#define NNSZ   ((size_t)N * (size_t)N)
#define NF     8
#define NCOEF  17          // order 16 + 1

// GEMM tiling: block tile 64(M) x 128(N), K-tile 32; 4 wave32s, each 32x64.
#define BM 64
#define BN 128
#define BK 32
#define LDA  (BK + 4)      // 36 floats  = 144B rows (16B aligned)
#define LDB2 (2 * BN + 4)  // 260 floats = 1040B rows (16B aligned), pair-interleaved

typedef __attribute__((ext_vector_type(2))) float v2f;
typedef __attribute__((ext_vector_type(4))) float vf4;
typedef __attribute__((ext_vector_type(8))) float v8f;

// T-buffer global layout is K-pair interleaved:
//   element (k, n) lives at (k>>1)*2N + 2n + (k&1)
// This makes the WMMA B-fragment (B[k][n], B[k+1][n]) an aligned 8-byte unit
// and keeps global<->LDS staging as contiguous float4s.
__device__ __forceinline__ size_t t_index(int row, int col) {
    return (size_t)(row >> 1) * N2 + (size_t)col * 2 + (row & 1);
}

// ---------------------------------------------------------------------------
// 1) Chebyshev coefficients: c[k*NF + f], k in [0,17), f in [0,8)
// ---------------------------------------------------------------------------
__global__ void cheby_coef_kernel(const float* __restrict__ taus,
                                  float* __restrict__ c) {
    int t = threadIdx.x;
    if (t >= NCOEF * NF) return;
    int k = t / NF;
    int f = t % NF;
    const float PI = 3.14159265358979323846f;
    float tau = taus[f];
    float sum = 0.0f;
    for (int n = 0; n < NCOEF; ++n) {
        float half = (float)n + 0.5f;
        float pt = cosf(PI * half / (float)NCOEF) + 1.0f;  // a1*cos + a2, a1=a2=1
        float fe = expf(-pt * tau);
        sum += cosf(PI * (float)k * half / (float)NCOEF) * fe;
    }
    c[t] = (2.0f / (float)NCOEF) * sum;
}

// ---------------------------------------------------------------------------
// 2) Init: F = 2(L - I) (row-major), T0 = I, T1 = L - I (pair-interleaved),
//    out[f] = 0.5*c[0,f]*I + c[1,f]*T1 (row-major)
// ---------------------------------------------------------------------------
__global__ void cheby_init_kernel(const float* __restrict__ L,
                                  const float* __restrict__ c,
                                  float* __restrict__ F,
                                  float* __restrict__ T0,
                                  float* __restrict__ T1,
                                  float* __restrict__ out) {
    size_t idx = (size_t)blockIdx.x * blockDim.x + threadIdx.x;
    int i = (int)(idx >> 11);          // / 2048
    int j = (int)(idx & (size_t)2047); // % 2048
    float l   = L[idx];
    float eye = (i == j) ? 1.0f : 0.0f;
    float t1  = l - eye;
    F[idx] = 2.0f * t1;
    size_t ti = t_index(i, j);
    T0[ti] = eye;
    T1[ti] = t1;
#pragma unroll
    for (int f = 0; f < NF; ++f) {
        out[(size_t)f * NNSZ + idx] = 0.5f * c[f] * eye + c[NF + f] * t1;
    }
}

// ---------------------------------------------------------------------------
// 3) One Chebyshev recurrence step, fused:
//    P = F @ Tcur (f32 WMMA GEMM); Tnew = P - Told; out[f] += ck[f] * Tnew
// ---------------------------------------------------------------------------
__global__ void __launch_bounds__(128)
cheby_step_kernel(const float* __restrict__ F,
                  const float* __restrict__ Tcur,   // pair-interleaved
                  const float* __restrict__ Told,   // pair-interleaved
                  float* __restrict__ Tnew,         // pair-interleaved
                  const float* __restrict__ ck,     // 8 coefficients for this k
                  float* __restrict__ out) {
    __shared__ __align__(16) float As[BM * LDA];        // 64 x 36
    __shared__ __align__(16) float Bs[(BK / 2) * LDB2]; // 16 x 260 (pair rows)

    const int tid   = threadIdx.x;
    const int lane  = tid & 31;
    const int wave  = tid >> 5;       // 0..3
    const int wm    = wave & 1;       // wave row (M dir): 2 x 32
    const int wn    = wave >> 1;      // wave col (N dir): 2 x 64
    const int bm    = blockIdx.y * BM;
    const int bn    = blockIdx.x * BN;

    const int lhalf = lane >> 4;      // 0/1: lane half selects K pair (A/B f32 frags)
    const int l15   = lane & 15;
    const int kbase = lhalf * 2;

    v8f acc[2][4];
#pragma unroll
    for (int i = 0; i < 2; ++i)
#pragma unroll
        for (int j = 0; j < 4; ++j)
#pragma unroll
            for (int r = 0; r < 8; ++r) acc[i][j][r] = 0.0f;

    // staging: A = 64x32 floats (4 float4/thread), B = 16 pair-rows x 256 floats
    // (8 float4/thread)
    const int ar  = tid >> 3;          // 0..15
    const int ac4 = (tid & 7) * 4;     // 0..28
    const int brow = tid >> 3;         // 0..15 (pair row)
    const int bc4  = (tid & 7) * 4;    // 0..28, stepped by 32 over 8 passes
    const size_t bn2 = (size_t)bn * 2;

    const int amRow0 = wm * 32 + l15;      // A fragment M (lanes carry M)
    const int nTile0 = wn * 64 + l15;      // B fragment N base (lanes carry N)

    for (int kt = 0; kt < N; kt += BK) {
        const int ktp = kt >> 1;  // pair-row base in interleaved Tcur
        // ---- global -> LDS (float4) ----
#pragma unroll
        for (int p = 0; p < 4; ++p) {
            int r = ar + 16 * p;
            vf4 v = *(const vf4*)(F + (size_t)(bm + r) * N + kt + ac4);
            *(vf4*)(As + r * LDA + ac4) = v;
        }
#pragma unroll
        for (int q = 0; q < 8; ++q) {
            int col = bc4 + 32 * q;   // 0..255 within pair row
            vf4 v = *(const vf4*)(Tcur + (size_t)(ktp + brow) * N2 + bn2 + col);
            *(vf4*)(Bs + brow * LDB2 + col) = v;
        }
        // prefetch next K-tile while this one is consumed
        if (kt + BK < N) {
            __builtin_prefetch(F + (size_t)(bm + ar) * N + (kt + BK) + ac4, 0, 3);
            __builtin_prefetch(Tcur + (size_t)(ktp + 16 + brow) * N2 + bn2 + bc4, 0, 3);
        }
        __syncthreads();

        // ---- 8 K-steps of 4; 8 WMMAs each ----
#pragma unroll
        for (int kk = 0; kk < BK; kk += 4) {
            v2f a[2], b[4];
#pragma unroll
            for (int mt = 0; mt < 2; ++mt)
                a[mt] = *(const v2f*)(As + (amRow0 + mt * 16) * LDA + kk + kbase);
            const int prow = (kk >> 1) + lhalf;   // pair row for K = kk+kbase
#pragma unroll
            for (int nt = 0; nt < 4; ++nt)
                b[nt] = *(const v2f*)(Bs + prow * LDB2 + (nTile0 + nt * 16) * 2);
#pragma unroll
            for (int mt = 0; mt < 2; ++mt)
#pragma unroll
                for (int nt = 0; nt < 4; ++nt)
                    acc[mt][nt] = __builtin_amdgcn_wmma_f32_16x16x4_f32(
                        false, a[mt], false, b[nt], (short)0, acc[mt][nt],
                        false, false);
        }
        __syncthreads();
    }

    // ---- fused epilogue: Tnew = P - Told ; out[f] += ck[f]*Tnew ----
    float cf[NF];
#pragma unroll
    for (int f = 0; f < NF; ++f) cf[f] = ck[f];

#pragma unroll
    for (int mt = 0; mt < 2; ++mt) {
#pragma unroll
        for (int nt = 0; nt < 4; ++nt) {
            // C/D layout: VGPR r -> M = r + lhalf*8; lanes carry N
            const int row0 = bm + wm * 32 + mt * 16 + lhalf * 8;
            const int col  = bn + wn * 64 + nt * 16 + l15;
#pragma unroll
            for (int r = 0; r < 8; ++r) {
                const int row = row0 + r;
                const size_t ti = t_index(row, col);
                float v = acc[mt][nt][r] - Told[ti];
                Tnew[ti] = v;
                const size_t oi = (size_t)row * N + col;
#pragma unroll
                for (int f = 0; f < NF; ++f) {
                    out[(size_t)f * NNSZ + oi] += cf[f] * v;
                }
            }
        }
    }
}

// ---------------------------------------------------------------------------
// Launch: deterministic, all on `stream`, graph-capture safe.
// Workspace (floats): [c:256][F:N*N][T0:N*N][T1:N*N][T2:N*N] ~= 64 MiB
// ---------------------------------------------------------------------------
extern "C" void kernel_launch(void* const* d_in, const int* in_sizes, int n_in,
                              void* d_out, int out_size, void* d_ws, size_t ws_size,
                              hipStream_t stream) {
    const float* L    = (const float*)d_in[0];
    const float* taus = (const float*)d_in[1];
    (void)in_sizes; (void)n_in; (void)out_size; (void)ws_size;

    float* ws = (float*)d_ws;
    float* c  = ws;                    // 136 used, 256 reserved
    float* F  = ws + 256;
    float* b0 = F  + NNSZ;
    float* b1 = b0 + NNSZ;
    float* b2 = b1 + NNSZ;
    float* out = (float*)d_out;

    cheby_coef_kernel<<<1, 160, 0, stream>>>(taus, c);
    cheby_init_kernel<<<(int)(NNSZ / 256), 256, 0, stream>>>(L, c, F, b0, b1, out);

    const float* told = b0;   // T_{k-1}
    const float* tcur = b1;   // T_k
    float*       tnew = b2;   // T_{k+1}
    dim3 grid(N / BN, N / BM);
    for (int k = 2; k < NCOEF; ++k) {
        cheby_step_kernel<<<grid, 128, 0, stream>>>(F, tcur, told, tnew,
                                                    c + k * NF, out);
        float* tmp = (float*)told;
        told = tcur;
        tcur = tnew;
        tnew = tmp;
    }
}